// TopKSAE_24111946400111
// MI455X (gfx1250) — compile-verified
//
#include <hip/hip_runtime.h>

// ---------------- problem constants ----------------
#define B_ROWS 4096
#define D_DIM  768
#define H_DIM  16384
#define TOPK   128

// ---------------- encoder GEMM tiling ----------------
#define TBM 128
#define TBN 128
#define TBK 16
#define LDSA 20    // A row stride (floats): 80B rows -> 16B-aligned b128 LDS writes
#define LDSB 136   // B row stride (floats): 544B rows -> 16B-aligned, half-waves in disjoint banks

typedef __attribute__((ext_vector_type(2))) float v2f;
typedef __attribute__((ext_vector_type(8))) float v8f;

__device__ __forceinline__ unsigned lds_off(const void* p) {
    // generic pointer to LDS: low 32 bits hold the addrspace(3) offset
    return (unsigned)(uintptr_t)p;
}

// ============================================================
// Kernel 0: enc_corr[h] = b_dec . W_enc[:,h]   (rank-1 fold of the centering)
// ============================================================
__global__ __launch_bounds__(256) void sae_enc_corr(
    const float* __restrict__ W_enc, const float* __restrict__ b_dec,
    float* __restrict__ corr)
{
    const int h = blockIdx.x * 256 + threadIdx.x;
    float s = 0.f;
    for (int k = 0; k < D_DIM; ++k)
        s = fmaf(b_dec[k], W_enc[(size_t)k * H_DIM + h], s);
    corr[h] = s;
}

// ============================================================
// Kernel 1: acts = relu(x @ W_enc - enc_corr), V_WMMA_F32_16X16X4_F32.
// 256 threads (8 waves), 128x128 block tile, K-step 16.
// Double-buffered LDS fed by GLOBAL_LOAD_ASYNC_TO_LDS_B128 (ASYNCcnt),
// so the next tile's DMA overlaps the current tile's 32 WMMAs.
// ============================================================
__global__ __launch_bounds__(256) void sae_encode_wmma(
    const float* __restrict__ x, const float* __restrict__ W_enc,
    const float* __restrict__ enc_corr, float* __restrict__ acts)
{
    __shared__ __align__(16) float lA[2][TBM * LDSA];   // [m][k]
    __shared__ __align__(16) float lB[2][TBK * LDSB];   // [k][n]

    const int tid   = threadIdx.x;
    const int lane  = tid & 31;
    const int wave  = tid >> 5;
    const int bm    = blockIdx.y * TBM;
    const int bn    = blockIdx.x * TBN;
    const int wm0   = (wave >> 1) * 32;   // 0,32,64,96
    const int wn0   = (wave & 1) * 64;    // 0,64
    const int l16   = lane & 15;
    const int lhalf = lane >> 4;
    const int kbase = lhalf * 2;

    // ---- per-thread async-copy slots (4 b128 lanes-worth per thread/tile) ----
    // A tile: 128 rows x 16 cols = 512 float4; slots tid and tid+256
    const int arow0 = tid >> 2,        ac0 = (tid & 3) << 2;
    const int arow1 = (tid + 256) >> 2, ac1 = (tid & 3) << 2;
    const float* gA0 = x + (size_t)(bm + arow0) * D_DIM + ac0;
    const float* gA1 = x + (size_t)(bm + arow1) * D_DIM + ac1;
    // B tile: 16 rows x 128 cols = 512 float4; krow = tid>>5 and +8
    const int bkr = tid >> 5, bn4 = (tid & 31) << 2;
    const float* gB0 = W_enc + (size_t)bkr * H_DIM + bn + bn4;
    const float* gB1 = gB0 + (size_t)8 * H_DIM;

    unsigned laA0[2], laA1[2], laB0[2], laB1[2];
#pragma unroll
    for (int b = 0; b < 2; ++b) {
        laA0[b] = lds_off(&lA[b][arow0 * LDSA + ac0]);
        laA1[b] = lds_off(&lA[b][arow1 * LDSA + ac1]);
        laB0[b] = lds_off(&lB[b][bkr * LDSB + bn4]);
        laB1[b] = lds_off(&lB[b][(bkr + 8) * LDSB + bn4]);
    }

    auto issue = [&](int k0, int buf) {
        asm volatile("global_load_async_to_lds_b128 %0, %1, off"
                     :: "v"(laA0[buf]), "v"(gA0 + k0) : "memory");
        asm volatile("global_load_async_to_lds_b128 %0, %1, off"
                     :: "v"(laA1[buf]), "v"(gA1 + k0) : "memory");
        asm volatile("global_load_async_to_lds_b128 %0, %1, off"
                     :: "v"(laB0[buf]), "v"(gB0 + (size_t)k0 * H_DIM) : "memory");
        asm volatile("global_load_async_to_lds_b128 %0, %1, off"
                     :: "v"(laB1[buf]), "v"(gB1 + (size_t)k0 * H_DIM) : "memory");
    };

    const v8f zero = {0.f,0.f,0.f,0.f,0.f,0.f,0.f,0.f};
    v8f acc[2][4];
#pragma unroll
    for (int i = 0; i < 2; ++i)
#pragma unroll
        for (int j = 0; j < 4; ++j) acc[i][j] = zero;

    issue(0, 0);
    int cur = 0;
    for (int k0 = 0; k0 < D_DIM; k0 += TBK) {
        if (k0 + TBK < D_DIM) {
            issue(k0 + TBK, cur ^ 1);
            // own 4 current-tile copies done when <=4 remain (in-order completion)
            asm volatile("s_wait_asynccnt 0x4" ::: "memory");
        } else {
            asm volatile("s_wait_asynccnt 0x0" ::: "memory");
        }
        __syncthreads();   // everyone's current buffer is resident

        const float* Ab = &lA[cur][0];
        const float* Bb = &lB[cur][0];
#pragma unroll
        for (int kk = 0; kk < TBK; kk += 4) {
            v2f af[2], bf[4];
#pragma unroll
            for (int i = 0; i < 2; ++i)
                af[i] = *(const v2f*)(Ab + (wm0 + i * 16 + l16) * LDSA + kk + kbase);
#pragma unroll
            for (int j = 0; j < 4; ++j) {
                const int nloc = wn0 + j * 16 + l16;
                bf[j].x = Bb[(kk + kbase) * LDSB + nloc];
                bf[j].y = Bb[(kk + kbase + 1) * LDSB + nloc];
            }
#pragma unroll
            for (int i = 0; i < 2; ++i)
#pragma unroll
                for (int j = 0; j < 4; ++j)
                    acc[i][j] = __builtin_amdgcn_wmma_f32_16x16x4_f32(
                        false, af[i], false, bf[j], (short)0, acc[i][j], false, false);
        }
        __syncthreads();   // all reads of `cur` done before it is re-filled
        cur ^= 1;
    }

    // ---- epilogue: subtract folded centering term, fused ReLU store ----
    float cc[4];
#pragma unroll
    for (int j = 0; j < 4; ++j)
        cc[j] = enc_corr[bn + wn0 + j * 16 + l16];

#pragma unroll
    for (int i = 0; i < 2; ++i) {
#pragma unroll
        for (int j = 0; j < 4; ++j) {
            const int col = bn + wn0 + j * 16 + l16;
#pragma unroll
            for (int r = 0; r < 8; ++r) {
                const int row = bm + wm0 + i * 16 + r + lhalf * 8;
                float v = acc[i][j][r] - cc[j];
                acts[(size_t)row * H_DIM + col] = v > 0.0f ? v : 0.0f;
            }
        }
    }
}

// ============================================================
// Kernel 2: exact per-row top-K=128 via 8-bit radix select on float bits
// (values >= 0 after ReLU -> uint order == float order). Keeps >T plus the
// first (K - count_gt) ties by index (jax top_k tie rule), zeroes the rest
// in place, writes compacted (idx,val) to workspace. 256 thr/row.
// ============================================================
__device__ __forceinline__ void scan8(bool pred, int lane, int wave,
                                      unsigned* swave, unsigned& excl, unsigned& total)
{
    unsigned bal  = __builtin_amdgcn_ballot_w32(pred);
    unsigned lpre = __builtin_popcount(bal & ((1u << lane) - 1u));
    if (lane == 0) swave[wave] = __builtin_popcount(bal);
    __syncthreads();
    unsigned wpre = 0, tot = 0;
#pragma unroll
    for (int w = 0; w < 8; ++w) {
        unsigned c = swave[w];
        wpre += (w < wave) ? c : 0u;
        tot  += c;
    }
    __syncthreads();
    excl = wpre + lpre;
    total = tot;
}

__global__ __launch_bounds__(256) void sae_topk(
    float* __restrict__ acts, int* __restrict__ ws_idx,
    float* __restrict__ ws_val, float* __restrict__ ws_l0)
{
    __shared__ unsigned hist[256];
    __shared__ unsigned swave[8];
    __shared__ unsigned bcast[2];

    const int tid  = threadIdx.x;
    const int lane = tid & 31;
    const int wave = tid >> 5;
    const int row  = blockIdx.x;
    float* rowp = acts + (size_t)row * H_DIM;
    const int PER = H_DIM / 256;   // 64

    unsigned prefix = 0u, prefmask = 0u, remaining = TOPK;

    for (int level = 3; level >= 0; --level) {
        hist[tid] = 0u;
        __syncthreads();
        const int shift = level * 8;
        for (int i = 0; i < PER; ++i) {
            unsigned u = __float_as_uint(rowp[(size_t)i * 256 + tid]);
            if ((u & prefmask) == prefix)
                atomicAdd(&hist[(u >> shift) & 0xFFu], 1u);
        }
        __syncthreads();
        if (tid == 0) {
            unsigned cum = 0u; unsigned chosen = 0u;
            for (int b = 255; b >= 0; --b) {
                unsigned c = hist[b];
                if (cum + c >= remaining) { chosen = (unsigned)b; break; }
                cum += c;
            }
            bcast[0] = chosen;
            bcast[1] = remaining - cum;   // still needed inside chosen bucket
        }
        __syncthreads();
        prefix   |= bcast[0] << shift;
        prefmask |= 0xFFu << shift;
        remaining = bcast[1];
        __syncthreads();
    }

    const float    T         = __uint_as_float(prefix);  // K-th largest value
    const unsigned need_ties = remaining;                // # of ==T to keep

    unsigned eq_run = 0, keep_run = 0;
    for (int i = 0; i < PER; ++i) {
        const int gidx = i * 256 + tid;
        const float v = rowp[gidx];
        const bool iseq = (v == T);
        unsigned eq_excl, eq_tot;
        scan8(iseq, lane, wave, swave, eq_excl, eq_tot);
        const bool keep = (v > T) || (iseq && (eq_run + eq_excl < need_ties));
        unsigned k_excl, k_tot;
        scan8(keep, lane, wave, swave, k_excl, k_tot);
        if (keep) {
            const unsigned kpos = keep_run + k_excl;
            ws_idx[(size_t)row * TOPK + kpos] = gidx;
            ws_val[(size_t)row * TOPK + kpos] = v;
        } else {
            rowp[gidx] = 0.0f;
        }
        eq_run   += eq_tot;
        keep_run += k_tot;
    }
    if (tid == 0)
        ws_l0[row] = (T > 0.0f) ? (float)TOPK : (float)(TOPK - need_ties);
}

// ============================================================
// Kernel 3: sparse decode x_rec = acts_topk @ W_dec + b_dec (128 nnz/row),
// fused with per-row squared-error partial sum. W_dec (48MB) is L2-resident.
// ============================================================
__global__ __launch_bounds__(256) void sae_decode(
    const float* __restrict__ x, const float* __restrict__ W_dec,
    const float* __restrict__ b_dec, const int* __restrict__ ws_idx,
    const float* __restrict__ ws_val, float* __restrict__ x_rec,
    float* __restrict__ ws_l2)
{
    __shared__ float sval[TOPK];
    __shared__ int   sidx[TOPK];
    __shared__ float red[256];

    const int tid = threadIdx.x;
    const int row = blockIdx.x;
    if (tid < TOPK) {
        sval[tid] = ws_val[(size_t)row * TOPK + tid];
        sidx[tid] = ws_idx[(size_t)row * TOPK + tid];
    }
    __syncthreads();

    float a0 = b_dec[tid], a1 = b_dec[tid + 256], a2 = b_dec[tid + 512];
#pragma unroll 4
    for (int k = 0; k < TOPK; ++k) {
        const float w = sval[k];
        const float* wd = W_dec + (size_t)sidx[k] * D_DIM;
        a0 = fmaf(w, wd[tid],       a0);
        a1 = fmaf(w, wd[tid + 256], a1);
        a2 = fmaf(w, wd[tid + 512], a2);
    }
    const float* xr = x + (size_t)row * D_DIM;
    float* out = x_rec + (size_t)row * D_DIM;
    const float e0 = a0 - xr[tid], e1 = a1 - xr[tid + 256], e2 = a2 - xr[tid + 512];
    out[tid] = a0; out[tid + 256] = a1; out[tid + 512] = a2;

    red[tid] = e0 * e0 + e1 * e1 + e2 * e2;
    __syncthreads();
#pragma unroll
    for (int off = 128; off >= 1; off >>= 1) {
        if (tid < off) red[tid] += red[tid + off];
        __syncthreads();
    }
    if (tid == 0) ws_l2[row] = red[0];
}

// ============================================================
// Kernel 4: finalize scalars (deterministic single-block reduction)
// ============================================================
__global__ __launch_bounds__(256) void sae_finalize(
    const float* __restrict__ ws_l2, const float* __restrict__ ws_l0,
    float* __restrict__ scalars)
{
    __shared__ float r2[256], r0[256];
    const int tid = threadIdx.x;
    float s2 = 0.f, s0 = 0.f;
    for (int r = tid; r < B_ROWS; r += 256) { s2 += ws_l2[r]; s0 += ws_l0[r]; }
    r2[tid] = s2; r0[tid] = s0;
    __syncthreads();
#pragma unroll
    for (int off = 128; off >= 1; off >>= 1) {
        if (tid < off) { r2[tid] += r2[tid + off]; r0[tid] += r0[tid + off]; }
        __syncthreads();
    }
    if (tid == 0) {
        const float l2 = r2[0] / (float)((size_t)B_ROWS * D_DIM);
        scalars[0] = l2;                    // loss (aux_loss == 0)
        scalars[1] = l2;                    // l2_loss
        scalars[2] = r0[0] / (float)B_ROWS; // l0_norm
    }
}

// ============================================================
extern "C" void kernel_launch(void* const* d_in, const int* in_sizes, int n_in,
                              void* d_out, int out_size, void* d_ws, size_t ws_size,
                              hipStream_t stream) {
    (void)in_sizes; (void)n_in; (void)out_size; (void)ws_size;
    const float* x     = (const float*)d_in[0];
    const float* W_enc = (const float*)d_in[1];
    const float* W_dec = (const float*)d_in[2];
    const float* b_dec = (const float*)d_in[3];

    float* x_rec   = (float*)d_out;
    float* acts    = x_rec + (size_t)B_ROWS * D_DIM;
    float* scalars = acts + (size_t)B_ROWS * H_DIM;

    // ws: vals[B*K] f32 | idx[B*K] i32 | l2[B] | l0[B] | corr[H]  (~4.2 MB)
    float* ws_val = (float*)d_ws;
    int*   ws_idx = (int*)((char*)d_ws + (size_t)B_ROWS * TOPK * sizeof(float));
    float* ws_l2  = (float*)((char*)d_ws + 2 * (size_t)B_ROWS * TOPK * sizeof(float));
    float* ws_l0  = ws_l2 + B_ROWS;
    float* corr   = ws_l0 + B_ROWS;

    sae_enc_corr<<<H_DIM / 256, 256, 0, stream>>>(W_enc, b_dec, corr);
    dim3 gE(H_DIM / TBN, B_ROWS / TBM);   // 128 x 32 blocks
    sae_encode_wmma<<<gE, 256, 0, stream>>>(x, W_enc, corr, acts);
    sae_topk<<<B_ROWS, 256, 0, stream>>>(acts, ws_idx, ws_val, ws_l0);
    sae_decode<<<B_ROWS, 256, 0, stream>>>(x, W_dec, b_dec, ws_idx, ws_val, x_rec, ws_l2);
    sae_finalize<<<1, 256, 0, stream>>>(ws_l2, ws_l0, scalars);
}